// FreeLoss_21096879357997
// MI455X (gfx1250) — compile-verified
//
#include <hip/hip_runtime.h>
#include <hip/hip_bf16.h>
#include <stdint.h>

// ---- problem constants (match reference) ----
#define B_    8
#define A_    32768
#define C_    80
#define M_    32
#define TOPK_ 50
#define VAR0_ 0.1f
#define VAR1_ 0.2f
#define T1_   0.5f
#define SL1W_ 0.75f
#define SL1B_ 0.11f

typedef __attribute__((ext_vector_type(2)))  float        v2f;
typedef __attribute__((ext_vector_type(8)))  float        v8f;
typedef __attribute__((ext_vector_type(4)))  unsigned int u32x4;
typedef __attribute__((ext_vector_type(8)))  int          i32x8;
typedef __attribute__((ext_vector_type(4)))  int          i32x4;

__device__ __forceinline__ float iou_corner(float ax0, float ay0, float ax1, float ay1,
                                            float bx0, float by0, float bx1, float by1) {
    float ix0 = fmaxf(ax0, bx0), iy0 = fmaxf(ay0, by0);
    float ix1 = fminf(ax1, bx1), iy1 = fminf(ay1, by1);
    float iw = fmaxf(ix1 - ix0, 0.f), ih = fmaxf(iy1 - iy0, 0.f);
    float inter = iw * ih;
    float areaA = (ax1 - ax0) * (ay1 - ay0);
    float areaB = (bx1 - bx0) * (by1 - by0);
    return inter / (areaA + areaB - inter);
}

// ======================= Kernel 1: per-(b,m) positive bag loss + t2 =======================
// One 1024-thread workgroup per (b,m). Each thread caches 32 IoU values in registers.
__global__ __launch_bounds__(1024) void k_pos(const float* __restrict__ box_reg,
                                              const float* __restrict__ cls_logits,
                                              const float* __restrict__ anchors,
                                              const float* __restrict__ tboxes,
                                              const int*   __restrict__ labels,
                                              float* __restrict__ t2_out,
                                              float* __restrict__ pos_out) {
    const int bm  = blockIdx.x;      // 0..B*M-1
    const int b   = bm / M_;
    const int m   = bm % M_;
    const int tid = threadIdx.x;
    const int lane = tid & 31;
    const int wid  = tid >> 5;

    __shared__ float              s_redf[32];
    __shared__ unsigned long long s_redu[32];
    __shared__ unsigned long long s_bcast;
    __shared__ int                s_sel[TOPK_];
    __shared__ float              s_w[TOPK_], s_wl[TOPK_];

    const float tx0 = tboxes[(b * M_ + m) * 4 + 0];
    const float ty0 = tboxes[(b * M_ + m) * 4 + 1];
    const float tx1 = tboxes[(b * M_ + m) * 4 + 2];
    const float ty1 = tboxes[(b * M_ + m) * 4 + 3];
    const int   lab = labels[b * M_ + m];

    float mqv[A_ / 1024];
    float obimax = 0.f;
#pragma unroll
    for (int j = 0; j < A_ / 1024; ++j) {
        const int a = tid + j * 1024;
        const float cx = anchors[a * 4 + 0], cy = anchors[a * 4 + 1];
        const float w  = anchors[a * 4 + 2], h  = anchors[a * 4 + 3];
        // point_form(anchor)
        mqv[j] = iou_corner(tx0, ty0, tx1, ty1,
                            cx - 0.5f * w, cy - 0.5f * h, cx + 0.5f * w, cy + 0.5f * h);
        // decoded = point_form(decode(box_reg, anchor))
        const float l0 = box_reg[(b * A_ + a) * 4 + 0];
        const float l1 = box_reg[(b * A_ + a) * 4 + 1];
        const float l2 = box_reg[(b * A_ + a) * 4 + 2];
        const float l3 = box_reg[(b * A_ + a) * 4 + 3];
        const float dcx = cx + l0 * VAR0_ * w;
        const float dcy = cy + l1 * VAR0_ * h;
        const float dw  = w * expf(l2 * VAR1_);
        const float dh  = h * expf(l3 * VAR1_);
        const float obi = iou_corner(tx0, ty0, tx1, ty1,
                                     dcx - 0.5f * dw, dcy - 0.5f * dh,
                                     dcx + 0.5f * dw, dcy + 0.5f * dh);
        obimax = fmaxf(obimax, obi);
    }

    // ---- t2 = max(max_a obi, T1+eps) ----
    for (int off = 16; off; off >>= 1) obimax = fmaxf(obimax, __shfl_xor(obimax, off));
    if (lane == 0) s_redf[wid] = obimax;
    __syncthreads();
    if (tid == 0) {
        float v = s_redf[0];
        for (int i = 1; i < 32; ++i) v = fmaxf(v, s_redf[i]);
        t2_out[bm] = fmaxf(v, (float)(0.5 + 1e-12));
    }

    // ---- exact top-50 of mq by iterative packed-u64 argmax ----
    // key = (float_bits(val) << 32) | (0xFFFFFFFF - idx): bigger value wins, tie -> smaller idx.
    unsigned long long last = ~0ull;
    for (int k = 0; k < TOPK_; ++k) {
        unsigned long long best = 0ull;
#pragma unroll
        for (int j = 0; j < A_ / 1024; ++j) {
            const unsigned vb = __float_as_uint(mqv[j]);
            const unsigned long long key =
                ((unsigned long long)vb << 32) | (unsigned long long)(0xFFFFFFFFu - (unsigned)(tid + j * 1024));
            if (key < last && key > best) best = key;
        }
        for (int off = 16; off; off >>= 1) {
            unsigned long long o = __shfl_xor(best, off);
            if (o > best) best = o;
        }
        if (lane == 0) s_redu[wid] = best;
        __syncthreads();
        if (tid == 0) {
            unsigned long long v = s_redu[0];
            for (int i = 1; i < 32; ++i) if (s_redu[i] > v) v = s_redu[i];
            s_bcast  = v;
            s_sel[k] = (int)(0xFFFFFFFFu - (unsigned)(v & 0xFFFFFFFFull));
        }
        __syncthreads();
        last = s_bcast;
    }

    // ---- bag loss over the 50 matched anchors ----
    if (tid < TOPK_) {
        const int aidx = s_sel[tid];
        const float pcx = anchors[aidx * 4 + 0], pcy = anchors[aidx * 4 + 1];
        const float pw  = anchors[aidx * 4 + 2], ph  = anchors[aidx * 4 + 3];
        // encode(target, prior)
        const float g0 = ((tx0 + tx1) * 0.5f - pcx) / (VAR0_ * pw);
        const float g1 = ((ty0 + ty1) * 0.5f - pcy) / (VAR0_ * ph);
        const float g2 = logf((tx1 - tx0) / pw) / VAR1_;
        const float g3 = logf((ty1 - ty0) / ph) / VAR1_;
        const float v0 = g0 - box_reg[(b * A_ + aidx) * 4 + 0];
        const float v1 = g1 - box_reg[(b * A_ + aidx) * 4 + 1];
        const float v2 = g2 - box_reg[(b * A_ + aidx) * 4 + 2];
        const float v3 = g3 - box_reg[(b * A_ + aidx) * 4 + 3];
        float reg = 0.f;
        {
            float av;
            av = fabsf(v0); reg += (av < SL1B_) ? (0.5f / SL1B_) * v0 * v0 : av - 0.5f * SL1B_;
            av = fabsf(v1); reg += (av < SL1B_) ? (0.5f / SL1B_) * v1 * v1 : av - 0.5f * SL1B_;
            av = fabsf(v2); reg += (av < SL1B_) ? (0.5f / SL1B_) * v2 * v2 : av - 0.5f * SL1B_;
            av = fabsf(v3); reg += (av < SL1B_) ? (0.5f / SL1B_) * v3 * v3 : av - 0.5f * SL1B_;
            reg *= SL1W_;
        }
        const float x  = cls_logits[(size_t)(b * A_ + aidx) * C_ + lab];
        const float mc = 1.f / (1.f + expf(-x));
        const float li = mc * expf(-reg);
        const float wr = 1.f / fmaxf(1.f - li, 1e-12f);
        s_w[tid]  = wr;
        s_wl[tid] = wr * li;
    }
    __syncthreads();
    if (tid == 0) {
        float s1 = 0.f, s2 = 0.f;
        for (int k = 0; k < TOPK_; ++k) { s1 += s_w[k]; s2 += s_wl[k]; }
        const float bag = s2 / s1;
        pos_out[bm] = -fmaxf(logf(fmaxf(bag, 1e-38f)), -100.f);
    }
}

// ======================= Kernel 2: negative (focal) loss =======================
// One thread per anchor; targets staged to LDS by the Tensor Data Mover (TDM).
struct NegSmem {
    float tgt[M_ * 4];   // MUST be first: TDM descriptor uses LDS offset 0
    float t2[M_];
    int   lab[M_];
    float red[256];
};

__global__ __launch_bounds__(256) void k_neg(const float* __restrict__ box_reg,
                                             const float* __restrict__ cls_logits,
                                             const float* __restrict__ anchors,
                                             const float* __restrict__ tboxes,
                                             const int*   __restrict__ labels,
                                             const float* __restrict__ t2_in,
                                             float* __restrict__ neg_out) {
    const int blk   = blockIdx.x;          // 0 .. B*(A/256)-1
    const int b     = blk / (A_ / 256);
    const int chunk = blk % (A_ / 256);
    const int tid   = threadIdx.x;

    __shared__ NegSmem sm;

    // ---- TDM: DMA the 32x4-float target tile for image b into LDS (offset 0) ----
    if (tid == 0) {
        const unsigned long long ga = (unsigned long long)(const void*)(tboxes + (size_t)b * M_ * 4);
        u32x4 g0;
        g0.x = 1u;                                   // count=1 (valid), gather off
        g0.y = 0u;                                   // lds_addr = 0 (sm.tgt)
        g0.z = (unsigned)(ga & 0xFFFFFFFFull);       // global_addr[31:0]
        g0.w = (unsigned)((ga >> 32) & 0x1FFFFFFull) // global_addr[56:32]
             | (2u << 30);                           // type = image
        i32x8 g1;
        g1[0] = (int)(2u << 16);                     // data_size = 4B
        g1[1] = (int)(128u << 16);                   // tensor_dim0 = 128 (low 16)
        g1[2] = (int)(1u << 16);                     // tensor_dim1 = 1
        g1[3] = (int)(128u << 16);                   // tile_dim0 = 128
        g1[4] = 0;                                   // tile_dim1/2 unused
        g1[5] = 128;                                 // tensor_dim0_stride = 128
        g1[6] = 0;
        g1[7] = 0;
        i32x4 g2; g2[0] = 0; g2[1] = 0; g2[2] = 0; g2[3] = 0;
        i32x4 g3; g3[0] = 0; g3[1] = 0; g3[2] = 0; g3[3] = 0;
        i32x8 g4; g4[0] = 0; g4[1] = 0; g4[2] = 0; g4[3] = 0;
                  g4[4] = 0; g4[5] = 0; g4[6] = 0; g4[7] = 0;
        __builtin_amdgcn_tensor_load_to_lds(g0, g1, g2, g3, g4, 0);
        __builtin_amdgcn_s_wait_tensorcnt(0);
    }
    if (tid < M_) {
        sm.t2[tid]  = t2_in[b * M_ + tid];
        sm.lab[tid] = labels[b * M_ + tid];
    }
    __syncthreads();

    const int a  = chunk * 256 + tid;
    const float l0 = box_reg[(b * A_ + a) * 4 + 0];
    const float l1 = box_reg[(b * A_ + a) * 4 + 1];
    const float l2 = box_reg[(b * A_ + a) * 4 + 2];
    const float l3 = box_reg[(b * A_ + a) * 4 + 3];
    const float cx = anchors[a * 4 + 0], cy = anchors[a * 4 + 1];
    const float w  = anchors[a * 4 + 2], h  = anchors[a * 4 + 3];
    const float dcx = cx + l0 * VAR0_ * w;
    const float dcy = cy + l1 * VAR0_ * h;
    const float dw  = w * expf(l2 * VAR1_);
    const float dh  = h * expf(l3 * VAR1_);
    const float dx0 = dcx - 0.5f * dw, dy0 = dcy - 0.5f * dh;
    const float dx1 = dcx + 0.5f * dw, dy1 = dcy + 0.5f * dh;

    float obp[M_];
    int   labr[M_];
#pragma unroll
    for (int mm = 0; mm < M_; ++mm) {
        const float q0 = sm.tgt[mm * 4 + 0], q1 = sm.tgt[mm * 4 + 1];
        const float q2 = sm.tgt[mm * 4 + 2], q3 = sm.tgt[mm * 4 + 3];
        const float obi = iou_corner(q0, q1, q2, q3, dx0, dy0, dx1, dy1);
        obp[mm]  = fminf(fmaxf((obi - T1_) / (sm.t2[mm] - T1_), 0.f), 1.f);
        labr[mm] = sm.lab[mm];
    }

    float acc = 0.f;
    const float* cl = cls_logits + (size_t)(b * A_ + a) * C_;
    for (int c = 0; c < C_; ++c) {
        float bp = 0.f;
#pragma unroll
        for (int mm = 0; mm < M_; ++mm)
            bp = (labr[mm] == c) ? fmaxf(bp, obp[mm]) : bp;
        const float x  = cl[c];
        const float pr = 1.f / (1.f + expf(-x));
        const float p  = pr * (1.f - bp);
        const float lg = fmaxf(logf(fmaxf(1.f - p, 1e-38f)), -100.f);
        acc += p * p * (-lg);
    }

    sm.red[tid] = acc;
    __syncthreads();
    for (int s = 128; s; s >>= 1) {
        if (tid < s) sm.red[tid] += sm.red[tid + s];
        __syncthreads();
    }
    if (tid == 0) neg_out[blk] = sm.red[0];
}

// ======================= Kernel 3: final reduction (WMMA f32 16x16x4) =======================
// 1 wave. Neg partials (1024 f32) summed via D += A(16x4-chunk) x ones(4x16); total = sum(D)/16.
__global__ __launch_bounds__(32) void k_final(const float* __restrict__ pos_in,
                                              const float* __restrict__ neg_in,
                                              float* __restrict__ out) {
    const int lane = threadIdx.x;
    v8f d = {};
    v2f ones; ones.x = 1.f; ones.y = 1.f;
    for (int t = 0; t < 16; ++t) {
        v2f av;
        av.x = neg_in[t * 64 + lane * 2 + 0];
        av.y = neg_in[t * 64 + lane * 2 + 1];
        d = __builtin_amdgcn_wmma_f32_16x16x4_f32(false, av, false, ones,
                                                  (short)0, d, false, false);
    }
    float s = d[0] + d[1] + d[2] + d[3] + d[4] + d[5] + d[6] + d[7];
    for (int off = 16; off; off >>= 1) s += __shfl_xor(s, off);
    const float negsum = s * (1.f / 16.f);

    float ps = 0.f;
    for (int i = lane; i < B_ * M_; i += 32) ps += pos_in[i];
    for (int off = 16; off; off >>= 1) ps += __shfl_xor(ps, off);

    if (lane == 0) {
        out[0] = (ps / (float)(B_ * M_)) * 0.5f;                    // positive_loss * ALPHA
        out[1] = (negsum / (float)(B_ * M_ * TOPK_)) * 0.5f;        // negative_loss * (1-ALPHA)
    }
}

extern "C" void kernel_launch(void* const* d_in, const int* in_sizes, int n_in,
                              void* d_out, int out_size, void* d_ws, size_t ws_size,
                              hipStream_t stream) {
    (void)in_sizes; (void)n_in; (void)out_size; (void)ws_size;
    const float* box_reg    = (const float*)d_in[0];   // [B,A,4]
    const float* cls_logits = (const float*)d_in[1];   // [B,A,C]
    const float* anchors    = (const float*)d_in[2];   // [A,4]
    const float* tboxes     = (const float*)d_in[3];   // [B,M,4]
    const int*   labels     = (const int*)d_in[4];     // [B,M]
    float* out = (float*)d_out;

    float* ws  = (float*)d_ws;
    float* t2  = ws;                 // B*M
    float* pos = ws + B_ * M_;       // B*M
    float* neg = ws + 2 * B_ * M_;   // B*(A/256) = 1024

    k_pos<<<B_ * M_, 1024, 0, stream>>>(box_reg, cls_logits, anchors, tboxes, labels, t2, pos);
    k_neg<<<B_ * (A_ / 256), 256, 0, stream>>>(box_reg, cls_logits, anchors, tboxes, labels, t2, neg);
    k_final<<<1, 32, 0, stream>>>(pos, neg, out);
}